// GATNet_24824910970943
// MI455X (gfx1250) — compile-verified
//
#include <hip/hip_runtime.h>
#include <math.h>

typedef __attribute__((ext_vector_type(2))) float v2f;
typedef __attribute__((ext_vector_type(8))) float v8f;

#define HEADS 8
#define DHEAD 16
#define HD    128   // H*D == F == 128

// ---------------- WMMA fp32 GEMM: C[MxNc] = A[MxK] @ B[KxNc] (+bias) --------
// One wave computes one 16x16 tile with V_WMMA_F32_16X16X4_F32, K step 4.
// OOB columns load a clamped (in-bounds) address; their lanes' results are
// simply not stored (a B column only feeds its own output column), so the
// inner loop has no guards: 1 b64 load + 2 b32 loads + wmma + pointer adds.
__global__ __launch_bounds__(256) void wmma_gemm_f32(
    const float* __restrict__ A, const float* __restrict__ B,
    const float* __restrict__ bias, float* __restrict__ C,
    int M, int K, int Nc, int nColTiles, int useBias) {
  const int wave = threadIdx.x >> 5;
  const int lane = threadIdx.x & 31;
  const int nRowTiles = (M + 15) >> 4;
  const int tile = blockIdx.x * 8 + wave;
  if (tile >= nRowTiles * nColTiles) return;   // wave-uniform: EXEC all-1 for WMMA
  const int rowTile = tile / nColTiles;
  const int colTile = tile - rowTile * nColTiles;
  const int m0 = rowTile << 4;
  const int n0 = colTile << 4;
  const int half = lane >> 4;      // 0: lanes 0-15, 1: lanes 16-31
  const int l    = lane & 15;
  const int col  = n0 + l;
  const bool colOK = (col < Nc);
  const int colC = min(col, Nc - 1);
  const int rA   = min(m0 + l, M - 1);

  // A 16x4 f32: lanes 0-15 hold K=k,k+1 ; lanes 16-31 hold K=k+2,k+3
  const float* __restrict__ aptr = A + (size_t)rA * K + 2 * half;
  // B 4x16 f32: lanes 0-15 rows k,k+1 ; lanes 16-31 rows k+2,k+3
  const float* __restrict__ bptr = B + (size_t)(2 * half) * Nc + colC;
  const size_t bstep = 4 * (size_t)Nc;

  v8f acc = {0.f, 0.f, 0.f, 0.f, 0.f, 0.f, 0.f, 0.f};
#pragma unroll 4
  for (int k = 0; k < K; k += 4) {
    v2f a = *(const v2f*)aptr;
    v2f b;
    b.x = bptr[0];
    b.y = bptr[Nc];
    aptr += 4;
    bptr += bstep;
    acc = __builtin_amdgcn_wmma_f32_16x16x4_f32(false, a, false, b, (short)0,
                                                acc, false, false);
  }
  if (colOK) {
    const float bv = useBias ? bias[col] : 0.f;
#pragma unroll
    for (int v = 0; v < 8; v++) {
      const int r = m0 + v + 8 * half;   // C layout: VGPR v -> row v (+8 hi half)
      if (r < M) C[(size_t)r * Nc + col] = acc[v] + bv;
    }
  }
}

// ---------------- per-(node,head) attention logits ---------------------------
__global__ __launch_bounds__(256) void attn_logits_kernel(
    const float* __restrict__ hx, const float* __restrict__ a_src,
    const float* __restrict__ a_dst, float* __restrict__ alS,
    float* __restrict__ alD, int Nn) {
  const int idx = blockIdx.x * blockDim.x + threadIdx.x;   // n*H + h
  if (idx >= Nn * HEADS) return;
  const int n = idx >> 3, h = idx & 7;
  const float4* hv = (const float4*)(hx + (size_t)n * HD + h * DHEAD);
  const float4* av = (const float4*)(a_src + h * DHEAD);
  const float4* bv = (const float4*)(a_dst + h * DHEAD);
  float s = 0.f, d = 0.f;
#pragma unroll
  for (int i = 0; i < 4; i++) {
    float4 xv = hv[i], aa = av[i], bb = bv[i];
    s += xv.x * aa.x + xv.y * aa.y + xv.z * aa.z + xv.w * aa.w;
    d += xv.x * bb.x + xv.y * bb.y + xv.z * bb.z + xv.w * bb.w;
  }
  alS[idx] = s;
  alD[idx] = d;
}

// ======================= CSR build (dst-sorted, built once) ==================
__device__ __forceinline__ void edge_sd(const int* __restrict__ ei, int e, int E,
                                        int& s, int& d) {
  if (e < E) { s = ei[e]; d = ei[E + e]; }
  else       { s = d = e - E; }                // synthesized self-loop
}

__global__ __launch_bounds__(256) void csr_count_kernel(
    const int* __restrict__ ei, int E, int Nn, int* __restrict__ counts) {
  const int e = blockIdx.x * blockDim.x + threadIdx.x;
  if (e >= E + Nn) return;
  int s, d; edge_sd(ei, e, E, s, d);
  atomicAdd(&counts[d], 1);
}

// single-workgroup chunked exclusive scan: offsets[0..Nn], offsets[Nn]=total
__global__ __launch_bounds__(1024) void csr_scan_kernel(
    const int* __restrict__ counts, int* __restrict__ offsets, int Nn) {
  __shared__ int buf[1024];
  __shared__ int carryS;
  const int tid = threadIdx.x;
  if (tid == 0) carryS = 0;
  __syncthreads();
  for (int base = 0; base < Nn; base += 1024) {
    const int i = base + tid;
    const int v = (i < Nn) ? counts[i] : 0;
    buf[tid] = v;
    __syncthreads();
#pragma unroll
    for (int off = 1; off < 1024; off <<= 1) {
      const int t = (tid >= off) ? buf[tid - off] : 0;
      __syncthreads();
      buf[tid] += t;
      __syncthreads();
    }
    const int incl  = buf[tid];
    const int carry = carryS;
    if (i < Nn) offsets[i] = carry + incl - v;   // exclusive
    __syncthreads();
    if (tid == 1023) carryS = carry + incl;
    __syncthreads();
  }
  if (tid == 0) offsets[Nn] = carryS;
}

__global__ __launch_bounds__(256) void csr_fill_kernel(
    const int* __restrict__ ei, int E, int Nn,
    const int* __restrict__ offsets, int* __restrict__ cursor,
    int* __restrict__ edgeSrc) {
  const int e = blockIdx.x * blockDim.x + threadIdx.x;
  if (e >= E + Nn) return;
  int s, d; edge_sd(ei, e, E, s, d);
  const int pos = atomicAdd(&cursor[d], 1);
  edgeSrc[offsets[d] + pos] = s;     // store src id directly: gather needs only this
}

// ---------------- segment softmax stats per (dst,head): pure gather ----------
__global__ __launch_bounds__(256) void mz_kernel(
    const int* __restrict__ offsets, const int* __restrict__ edgeSrc,
    const float* __restrict__ alS, const float* __restrict__ alD,
    float* __restrict__ m, float* __restrict__ z, int Nn) {
  const int idx = blockIdx.x * blockDim.x + threadIdx.x;   // d*H + h
  if (idx >= Nn * HEADS) return;
  const int d = idx >> 3, h = idx & 7;
  const int beg = offsets[d], end = offsets[d + 1];
  const float ald = alD[idx];
  float mx = -3.4028235e38f;
  for (int j = beg; j < end; j++) {
    float v = alS[edgeSrc[j] * HEADS + h] + ald;
    v = v > 0.f ? v : 0.2f * v;                  // leaky_relu(0.2)
    mx = fmaxf(mx, v);
  }
  float sum = 0.f;
  for (int j = beg; j < end; j++) {
    float v = alS[edgeSrc[j] * HEADS + h] + ald;
    v = v > 0.f ? v : 0.2f * v;
    sum += expf(v - mx);
  }
  m[idx] = mx;
  z[idx] = (end > beg) ? sum : 1.f;
}

// ---------------- gather-aggregate: one wave per dst node --------------------
// lane l owns features 4l..4l+3 (head h = l/4). Per edge: one wave-uniform
// edgeSrc read + one L2-resident 512B hx row; accumulate in registers.
// Bias and (optionally) ELU fused into the epilogue. Zero atomics.
__global__ __launch_bounds__(256) void gat_aggregate_kernel(
    const int* __restrict__ offsets, const int* __restrict__ edgeSrc,
    const float* __restrict__ alS, const float* __restrict__ alD,
    const float* __restrict__ m, const float* __restrict__ z,
    const float* __restrict__ hx, const float* __restrict__ bias,
    float* __restrict__ out, int Nn, int doElu) {
  const int d = (blockIdx.x * blockDim.x + threadIdx.x) >> 5;   // dst node
  const int lane = threadIdx.x & 31;
  if (d >= Nn) return;                                           // wave-uniform
  const int h = lane >> 2;
  const float mh  = m[d * HEADS + h];
  const float rz  = 1.f / z[d * HEADS + h];
  const float ald = alD[d * HEADS + h];
  const int beg = offsets[d], end = offsets[d + 1];
  float4 acc = {0.f, 0.f, 0.f, 0.f};
  int sNext = (beg < end) ? edgeSrc[beg] : 0;
  for (int j = beg; j < end; j++) {
    const int s = sNext;
    if (j + 1 < end) {
      sNext = edgeSrc[j + 1];
      __builtin_prefetch(hx + (size_t)sNext * HD + lane * 4, 0, 3);
    }
    float v = alS[s * HEADS + h] + ald;
    v = v > 0.f ? v : 0.2f * v;
    const float w = expf(v - mh) * rz;
    const float4 hv = *(const float4*)(hx + (size_t)s * HD + lane * 4);
    acc.x += w * hv.x; acc.y += w * hv.y; acc.z += w * hv.z; acc.w += w * hv.w;
  }
  const float4 bb = *(const float4*)(bias + lane * 4);
  acc.x += bb.x; acc.y += bb.y; acc.z += bb.z; acc.w += bb.w;
  if (doElu) {
    acc.x = acc.x > 0.f ? acc.x : expm1f(acc.x);
    acc.y = acc.y > 0.f ? acc.y : expm1f(acc.y);
    acc.z = acc.z > 0.f ? acc.z : expm1f(acc.z);
    acc.w = acc.w > 0.f ? acc.w : expm1f(acc.w);
  }
  *(float4*)(out + (size_t)d * HD + lane * 4) = acc;
}

extern "C" void kernel_launch(void* const* d_in, const int* in_sizes, int n_in,
                              void* d_out, int out_size, void* d_ws, size_t ws_size,
                              hipStream_t stream) {
  const float* x   = (const float*)d_in[0];
  const int*   ei  = (const int*)d_in[1];
  const float* W1  = (const float*)d_in[2];
  const float* aS1 = (const float*)d_in[3];
  const float* aD1 = (const float*)d_in[4];
  const float* b1  = (const float*)d_in[5];
  const float* W2  = (const float*)d_in[6];
  const float* aS2 = (const float*)d_in[7];
  const float* aD2 = (const float*)d_in[8];
  const float* b2  = (const float*)d_in[9];
  const float* Wc  = (const float*)d_in[10];
  const float* bc  = (const float*)d_in[11];
  float* out = (float*)d_out;

  const int N = in_sizes[0] / HD;
  const int E = in_sizes[1] / 2;
  const int C = in_sizes[11];
  const int Etot = E + N;

  const size_t nF = (size_t)N * HD;
  const size_t nH = (size_t)N * HEADS;
  float* bufA = (float*)d_ws;          // hx of current layer
  float* bufB = bufA + nF;             // layer output (post bias/act)
  float* alS  = bufB + nF;
  float* alD  = alS + nH;
  float* m    = alD + nH;
  float* z    = m + nH;
  int* counts  = (int*)(z + nH);       // also reused as fill cursor
  int* offsets = counts + N;           // N+1
  int* edgeSrc = offsets + (N + 1);    // Etot

  const dim3 blk(256);
  const int gemmBlocks128 = (((N + 15) / 16) * (HD / 16) + 7) / 8;
  const int nhBlocks   = (int)((nH + 255) / 256);
  const int edgeBlocks = (Etot + 255) / 256;
  const int aggBlocks  = (N * 32 + 255) / 256;   // one wave per dst node

  // ---- build dst-CSR once (edge_index is layer-invariant) ----
  hipMemsetAsync(counts, 0, (size_t)N * sizeof(int), stream);
  csr_count_kernel<<<edgeBlocks, blk, 0, stream>>>(ei, E, N, counts);
  csr_scan_kernel<<<1, 1024, 0, stream>>>(counts, offsets, N);
  hipMemsetAsync(counts, 0, (size_t)N * sizeof(int), stream);
  csr_fill_kernel<<<edgeBlocks, blk, 0, stream>>>(ei, E, N, offsets, counts,
                                                  edgeSrc);

  for (int layer = 0; layer < 2; layer++) {
    const float* Ain = (layer == 0) ? x   : bufB;
    const float* W   = (layer == 0) ? W1  : W2;
    const float* aS  = (layer == 0) ? aS1 : aS2;
    const float* aD  = (layer == 0) ? aD1 : aD2;
    const float* bb  = (layer == 0) ? b1  : b2;

    // hx = Ain @ W   (fp32 WMMA)
    wmma_gemm_f32<<<gemmBlocks128, blk, 0, stream>>>(Ain, W, nullptr, bufA,
                                                     N, HD, HD, HD / 16, 0);
    attn_logits_kernel<<<nhBlocks, blk, 0, stream>>>(bufA, aS, aD, alS, alD, N);
    mz_kernel<<<nhBlocks, blk, 0, stream>>>(offsets, edgeSrc, alS, alD, m, z, N);
    gat_aggregate_kernel<<<aggBlocks, blk, 0, stream>>>(
        offsets, edgeSrc, alS, alD, m, z, bufA, bb, bufB, N,
        layer == 0 ? 1 : 0);
  }

  // classifier: out = bufB @ Wc + bc   (Nc = 40 -> 3 col tiles, store-guarded)
  const int cTiles = (C + 15) / 16;
  const int gemmBlocksC = (((N + 15) / 16) * cTiles + 7) / 8;
  wmma_gemm_f32<<<gemmBlocksC, blk, 0, stream>>>(bufB, Wc, bc, out,
                                                 N, HD, C, cTiles, 1);
}